// _CausalSPM_51015621541947
// MI455X (gfx1250) — compile-verified
//
#include <hip/hip_runtime.h>
#include <hip/hip_bf16.h>
#include <math.h>

typedef __attribute__((ext_vector_type(2))) float v2f;
typedef __attribute__((ext_vector_type(8))) float v8f;

#define B_  2
#define C_  256
#define HH  512
#define WW  512
#define NR  16
#define NC  16
#define SP  256   // NR*NC
#define PP  32    // pool size

// ---------------------------------------------------------------------------
// Kernel 1: 32x32 block-mean pool.  Streams the 536MB input with float4 loads.
// One 256-thread block per (b,c,nr,nc) tile: 1024 floats -> 256 float4 loads.
// ---------------------------------------------------------------------------
__global__ void k_pool(const float* __restrict__ h, float* __restrict__ pooled) {
    __shared__ float s[256];
    const int id = blockIdx.x;                 // B*C*NR*NC = 131072 blocks
    const int nc = id & 15;
    const int nr = (id >> 4) & 15;
    const int c  = (id >> 8) & 255;
    const int b  = id >> 16;
    const int t  = threadIdx.x;
    const int r  = t >> 3;                     // 0..31 row in tile
    const int q  = t & 7;                      // 0..7 float4 within row

    const size_t base = (((size_t)(b * C_ + c)) * HH + (nr * PP + r)) * WW
                        + (size_t)(nc * PP + q * 4);
    const float4 v = *reinterpret_cast<const float4*>(h + base);
    s[t] = v.x + v.y + v.z + v.w;
    __syncthreads();
    for (int off = 128; off > 0; off >>= 1) {
        if (t < off) s[t] += s[t + off];
        __syncthreads();
    }
    if (t == 0)
        pooled[(size_t)(b * C_ + c) * SP + nr * NC + nc] = s[0] * (1.0f / (PP * PP));
}

// ---------------------------------------------------------------------------
// Kernel 2: channel norm (mean/var over C per spatial position) + scale/shift.
// One block per (b, p); thread == channel.
// ---------------------------------------------------------------------------
__global__ void k_norm(const float* __restrict__ pooled,
                       const float* __restrict__ scale,
                       const float* __restrict__ shift,
                       float* __restrict__ xn) {
    __shared__ float s1[256];
    __shared__ float s2[256];
    const int p = blockIdx.x & (SP - 1);
    const int b = blockIdx.x >> 8;
    const int c = threadIdx.x;
    const float v = pooled[(size_t)(b * C_ + c) * SP + p];
    s1[c] = v;
    s2[c] = v * v;
    __syncthreads();
    for (int off = 128; off > 0; off >>= 1) {
        if (c < off) { s1[c] += s1[c + off]; s2[c] += s2[c + off]; }
        __syncthreads();
    }
    const float mean = s1[0] * (1.0f / C_);
    const float var  = s2[0] * (1.0f / C_) - mean * mean;
    const float inv  = rsqrtf(var + 1e-5f);
    xn[(size_t)(b * C_ + c) * SP + p] = (v - mean) * inv * scale[c] + shift[c];
}

// ---------------------------------------------------------------------------
// Kernel 3: depthwise 3x3 (pad=1) + bias + exact GELU on the 16x16 grid.
// ---------------------------------------------------------------------------
__global__ void k_dwgelu(const float* __restrict__ xn,
                         const float* __restrict__ dw_w,
                         const float* __restrict__ dw_b,
                         float* __restrict__ g) {
    const int idx = blockIdx.x * 256 + threadIdx.x;     // B*C*SP = 131072
    const int p = idx & (SP - 1);
    const int c = (idx >> 8) & 255;
    const int b = idx >> 16;
    const int nr = p >> 4, nc = p & 15;
    float acc = dw_b[c];
    #pragma unroll
    for (int dr = 0; dr < 3; ++dr) {
        #pragma unroll
        for (int dc = 0; dc < 3; ++dc) {
            const int rr = nr + dr - 1, cc = nc + dc - 1;
            if (rr >= 0 && rr < NR && cc >= 0 && cc < NC)
                acc += xn[(size_t)(b * C_ + c) * SP + rr * NC + cc]
                       * dw_w[c * 9 + dr * 3 + dc];
        }
    }
    // exact GELU: 0.5*x*(1+erf(x/sqrt(2)))
    const float ge = 0.5f * acc * (1.0f + erff(acc * 0.70710678118654752f));
    g[(size_t)(b * C_ + c) * SP + p] = ge;
}

// ---------------------------------------------------------------------------
// Kernel 4: pointwise 1x1 conv as fp32 WMMA GEMM.
// Per batch: D[o,p] = sum_c W[o,c]*G[c,p] + bias[o]  (256x256x256).
// One wave per 16x16 output tile; K stepped by 4 with V_WMMA_F32_16X16X4_F32.
// A layout: lanes0-15 hold K={k,k+1}, lanes16-31 hold K={k+2,k+3} (float2).
// B layout mirrors with N across lanes. C/D: v[j] -> (M=j+8*half, N=lane&15).
// 8 waves per 256-thread block; uniform control flow keeps EXEC all-1s.
// ---------------------------------------------------------------------------
__global__ void k_pwgemm(const float* __restrict__ pw_w,
                         const float* __restrict__ pw_b,
                         const float* __restrict__ g,
                         float* __restrict__ dsmall) {
    const int wave = blockIdx.x * 8 + (threadIdx.x >> 5);  // 512 tiles total
    const int b  = wave >> 8;
    const int tl = wave & 255;
    const int m0 = (tl >> 4) << 4;     // output-channel tile base
    const int n0 = (tl & 15) << 4;     // spatial tile base
    const int lane = threadIdx.x & 31;
    const int half = lane >> 4;        // 0: K=k,k+1  1: K=k+2,k+3
    const int li   = lane & 15;

    const float* gb = g + (size_t)b * C_ * SP;

    v8f acc = {};
    const int mrow = m0 + li;          // A row (output channel)
    const int ncol = n0 + li;          // B col (spatial position)
    for (int k = 0; k < C_; k += 4) {
        const int kk = k + half * 2;
        // A: W[mrow][kk], W[mrow][kk+1]  (contiguous, 8B aligned)
        const v2f a = *reinterpret_cast<const v2f*>(pw_w + (size_t)mrow * C_ + kk);
        // B: G[kk][ncol], G[kk+1][ncol]  (stride SP)
        v2f bb;
        bb.x = gb[(size_t)kk * SP + ncol];
        bb.y = gb[(size_t)(kk + 1) * SP + ncol];
        acc = __builtin_amdgcn_wmma_f32_16x16x4_f32(
            /*neg_a=*/false, a, /*neg_b=*/false, bb,
            /*c_mod=*/(short)0, acc, /*reuse_a=*/false, /*reuse_b=*/false);
    }
    float* db = dsmall + (size_t)b * C_ * SP;
    #pragma unroll
    for (int j = 0; j < 8; ++j) {
        const int m = m0 + half * 8 + j;
        db[(size_t)m * SP + n0 + li] = acc[j] + pw_b[m];
    }
}

// ---------------------------------------------------------------------------
// Kernel 5: broadcast upsample 16x16 -> 512x512, float4 stores (536MB write).
// ---------------------------------------------------------------------------
__global__ void k_upsample(const float* __restrict__ dsmall,
                           float4* __restrict__ out) {
    const int idx = blockIdx.x * 256 + threadIdx.x;   // 33554432 float4 stores
    const int w4 = idx & 127;            // W/4 = 128
    int rest = idx >> 7;
    const int hh = rest & 511;
    rest >>= 9;
    const int c = rest & 255;
    const int b = rest >> 8;
    const float v = dsmall[(size_t)(b * C_ + c) * SP + (hh >> 5) * NC + (w4 >> 3)];
    out[idx] = make_float4(v, v, v, v);
}

// ---------------------------------------------------------------------------
extern "C" void kernel_launch(void* const* d_in, const int* in_sizes, int n_in,
                              void* d_out, int out_size, void* d_ws, size_t ws_size,
                              hipStream_t stream) {
    const float* h_past = (const float*)d_in[0];
    const float* scale  = (const float*)d_in[1];
    const float* shift  = (const float*)d_in[2];
    const float* dw_w   = (const float*)d_in[3];
    const float* dw_b   = (const float*)d_in[4];
    const float* pw_w   = (const float*)d_in[5];
    const float* pw_b   = (const float*)d_in[6];

    const size_t small_elems = (size_t)B_ * C_ * SP;   // 131072 floats = 512KB
    float* pooled = (float*)d_ws;
    float* xn     = pooled + small_elems;
    float* g      = xn + small_elems;
    float* dsmall = g + small_elems;

    // 1) block-mean pool (bandwidth: 536MB read)
    k_pool<<<B_ * C_ * NR * NC, 256, 0, stream>>>(h_past, pooled);
    // 2) channel norm + affine
    k_norm<<<B_ * SP, 256, 0, stream>>>(pooled, scale, shift, xn);
    // 3) depthwise 3x3 + GELU
    k_dwgelu<<<(B_ * C_ * SP) / 256, 256, 0, stream>>>(xn, dw_w, dw_b, g);
    // 4) pointwise 1x1 conv via fp32 WMMA (512 tiles, 8 waves/block)
    k_pwgemm<<<64, 256, 0, stream>>>(pw_w, pw_b, g, dsmall);
    // 5) broadcast upsample (bandwidth: 536MB write)
    k_upsample<<<(B_ * C_ * HH * (WW / 4)) / 256, 256, 0, stream>>>(
        dsmall, (float4*)d_out);
}